// hierarchical_mutual_attn_85074712199520
// MI455X (gfx1250) — compile-verified
//
#include <hip/hip_runtime.h>
#include <math.h>

// Problem constants fixed by the reference.
#define D        128           // hidden dim
#define BGRAPHS  256           // num_graphs
#define LS       (D + 2)       // padded LDS row stride (floats) -> conflict-free
#define NTHREADS 256           // 8 wave32 waves
#define NWAVES   8

typedef float v2f __attribute__((ext_vector_type(2)));
typedef float v8f __attribute__((ext_vector_type(8)));

// D(16x16,f32) += A(16x4,f32) * B(4x16,f32); wave32, K=4 per issue.
__device__ __forceinline__ v8f wmma_f32_k4(v2f a, v2f b, v8f c) {
  return __builtin_amdgcn_wmma_f32_16x16x4_f32(
      /*neg_a=*/false, a, /*neg_b=*/false, b,
      /*c_mod=*/(short)0, c, /*reuse_a=*/false, /*reuse_b=*/false);
}

// dst[r, c] = sum_k sX[r, k] * W[c, k] + bias[c]   (i.e. x @ W^T + b)
// sX: LDS, Rp x D at stride LS (rows >= R are zero-padded).
// W : global, row-major D x D.  B-operand element B[k][n] = W[col0+n][k].
__device__ __forceinline__ void project_block(
    const float* __restrict__ sX, int Rp,
    const float* __restrict__ W, const float* __restrict__ bias,
    float* __restrict__ dst, int wave, int lane)
{
  const int nCT   = D / 16;                 // 8 column tiles
  const int tiles = (Rp >> 4) * nCT;
  const int hi = lane >> 4;                 // 0: K=0,1  1: K=2,3
  const int ln = lane & 15;
  for (int t = wave; t < tiles; t += NWAVES) {
    const int row0 = (t / nCT) << 4;
    const int col0 = (t % nCT) << 4;
    const float* aP = sX + (row0 + ln) * LS + 2 * hi;
    const float* bP = W + (size_t)(col0 + ln) * D + 2 * hi;
    v8f acc = {0.f, 0.f, 0.f, 0.f, 0.f, 0.f, 0.f, 0.f};
#pragma unroll
    for (int k0 = 0; k0 < D; k0 += 4) {
      v2f a = *(const v2f*)(aP + k0);       // LDS, 8B aligned
      v2f b = *(const v2f*)(bP + k0);       // global (L2-resident weights)
      acc = wmma_f32_k4(a, b, acc);
    }
    const float bv = bias[col0 + ln];
#pragma unroll
    for (int v = 0; v < 8; ++v)             // D layout: VGPR v -> M = v + 8*hi
      dst[(row0 + v + 8 * hi) * LS + (col0 + ln)] = acc[v] + bv;
  }
}

// One workgroup per graph: fused projections + block mutual attention +
// segment softmax + weighted pooling. Exploits that only the block-diagonal
// of alpha is ever consumed by the reference.
__global__ void __launch_bounds__(NTHREADS)
mutual_attn_kernel(const float* __restrict__ Xh, const float* __restrict__ Xt,
                   const float* __restrict__ Whw, const float* __restrict__ Whb,
                   const float* __restrict__ Wtw, const float* __restrict__ Wtb,
                   float* __restrict__ out_sh, float* __restrict__ out_st,
                   int Rh, int Rt, int RpH, int RpT)
{
  extern __shared__ float smem[];
  float* sXh = smem;                       // RpH x LS
  float* sXt = sXh + RpH * LS;             // RpT x LS
  float* ph1 = sXt + RpT * LS;             // Xh @ Wh^T + bh
  float* ph2 = ph1 + RpH * LS;             // Xh @ Wt^T + bt
  float* pt1 = ph2 + RpH * LS;             // Xt @ Wt^T + bt
  float* pt2 = pt1 + RpT * LS;             // Xt @ Wh^T + bh
  const int SLS = RpT + 2;
  float* sS  = pt2 + RpT * LS;             // RpH x SLS : tanh scores
  float* sbt = sS + RpH * SLS;             // Rt  pooled col-means
  float* spt = sbt + RpT;                  // Rt  softmax weights
  float* sbh = spt + RpT;                  // Rh  pooled row-means
  float* sph = sbh + RpH;                  // Rh  softmax weights

  const int tid  = threadIdx.x;
  const int wave = tid >> 5;
  const int lane = tid & 31;
  const int g    = blockIdx.x;

  // ---- Stage 1: stage this graph's node blocks into LDS (zero-padded) ----
  for (int idx = tid; idx < RpH * D; idx += NTHREADS) {
    int r = idx / D, c = idx - r * D;
    sXh[r * LS + c] = (r < Rh) ? Xh[(size_t)(g * Rh + r) * D + c] : 0.f;
  }
  for (int idx = tid; idx < RpT * D; idx += NTHREADS) {
    int r = idx / D, c = idx - r * D;
    sXt[r * LS + c] = (r < Rt) ? Xt[(size_t)(g * Rt + r) * D + c] : 0.f;
  }
  __syncthreads();

  // ---- Stage 2: four linear projections via f32 WMMA ----
  project_block(sXh, RpH, Whw, Whb, ph1, wave, lane);
  project_block(sXh, RpH, Wtw, Wtb, ph2, wave, lane);
  project_block(sXt, RpT, Wtw, Wtb, pt1, wave, lane);
  project_block(sXt, RpT, Whw, Whb, pt2, wave, lane);
  __syncthreads();

  // ---- Stage 3: S = ph1*pt1^T + ph2*pt2^T ; alpha = tanh(S) ----
  {
    const int nCT   = RpT >> 4;
    const int tiles = (RpH >> 4) * nCT;
    const int hi = lane >> 4, ln = lane & 15;
    for (int t = wave; t < tiles; t += NWAVES) {
      const int row0 = (t / nCT) << 4;
      const int col0 = (t % nCT) << 4;
      const float* a1 = ph1 + (row0 + ln) * LS + 2 * hi;
      const float* b1 = pt1 + (col0 + ln) * LS + 2 * hi;   // B[k][n] = pt1[n][k]
      const float* a2 = ph2 + (row0 + ln) * LS + 2 * hi;
      const float* b2 = pt2 + (col0 + ln) * LS + 2 * hi;
      v8f acc = {0.f, 0.f, 0.f, 0.f, 0.f, 0.f, 0.f, 0.f};
#pragma unroll
      for (int k0 = 0; k0 < D; k0 += 4)
        acc = wmma_f32_k4(*(const v2f*)(a1 + k0), *(const v2f*)(b1 + k0), acc);
#pragma unroll
      for (int k0 = 0; k0 < D; k0 += 4)
        acc = wmma_f32_k4(*(const v2f*)(a2 + k0), *(const v2f*)(b2 + k0), acc);
#pragma unroll
      for (int v = 0; v < 8; ++v)
        sS[(row0 + v + 8 * hi) * SLS + (col0 + ln)] = tanhf(acc[v]);
    }
  }
  __syncthreads();

  // ---- Stage 4: pooled means (divisors = exact segment counts) ----
  for (int j = tid; j < Rt; j += NTHREADS) {
    float s = 0.f;
    for (int i = 0; i < Rh; ++i) s += sS[i * SLS + j];
    sbt[j] = s / (float)Rh;
  }
  for (int i = tid; i < Rh; i += NTHREADS) {
    float s = 0.f;
    for (int j = 0; j < Rt; ++j) s += sS[i * SLS + j];
    sbh[i] = s / (float)Rt;
  }
  __syncthreads();

  // ---- Stage 5: per-graph softmax (segment is this whole block) ----
  if (tid == 0) {
    float m = -3.402823466e38f;
    for (int j = 0; j < Rt; ++j) m = fmaxf(m, sbt[j]);
    float s = 0.f;
    for (int j = 0; j < Rt; ++j) { float e = __expf(sbt[j] - m); spt[j] = e; s += e; }
    float inv = 1.f / s;
    for (int j = 0; j < Rt; ++j) spt[j] *= inv;
  } else if (tid == 32) {
    float m = -3.402823466e38f;
    for (int i = 0; i < Rh; ++i) m = fmaxf(m, sbh[i]);
    float s = 0.f;
    for (int i = 0; i < Rh; ++i) { float e = __expf(sbh[i] - m); sph[i] = e; s += e; }
    float inv = 1.f / s;
    for (int i = 0; i < Rh; ++i) sph[i] *= inv;
  }
  __syncthreads();

  // ---- Stage 6: weighted pooling of ORIGINAL features -> outputs ----
  if (tid < D) {
    const int d = tid;
    float acc = 0.f;
    for (int j = 0; j < Rt; ++j) acc = fmaf(spt[j], sXt[j * LS + d], acc);
    out_st[(size_t)g * D + d] = acc;
  } else {
    const int d = tid - D;
    float acc = 0.f;
    for (int i = 0; i < Rh; ++i) acc = fmaf(sph[i], sXh[i * LS + d], acc);
    out_sh[(size_t)g * D + d] = acc;
  }
}

static inline size_t lds_bytes(int RpH, int RpT) {
  size_t fl = (size_t)3 * RpH * LS + (size_t)3 * RpT * LS   // X + 2 projections/side
            + (size_t)RpH * (RpT + 2)                        // score block
            + 2u * (RpH + RpT) + 8;                          // reduction scratch
  return fl * sizeof(float);
}

extern "C" void kernel_launch(void* const* d_in, const int* in_sizes, int n_in,
                              void* d_out, int out_size, void* d_ws, size_t ws_size,
                              hipStream_t stream) {
  (void)n_in; (void)d_ws; (void)ws_size; (void)out_size;
  const float* xnh  = (const float*)d_in[0];
  const float* xnt  = (const float*)d_in[1];
  const float* xlh  = (const float*)d_in[2];
  const float* xlt  = (const float*)d_in[3];
  const float* Whw  = (const float*)d_in[4];
  const float* Whb  = (const float*)d_in[5];
  const float* Wtw  = (const float*)d_in[6];
  const float* Wtb  = (const float*)d_in[7];
  const float* Wlhw = (const float*)d_in[8];
  const float* Wlhb = (const float*)d_in[9];
  const float* Wltw = (const float*)d_in[10];
  const float* Wltb = (const float*)d_in[11];
  float* out = (float*)d_out;

  const int B  = BGRAPHS;
  const int Nh = in_sizes[0] / D, Nt = in_sizes[1] / D;
  const int Eh = in_sizes[2] / D, Et = in_sizes[3] / D;
  const int Rh  = Nh / B, Rt  = Nt / B;    // 24, 24
  const int Reh = Eh / B, Ret = Et / B;    // 40, 40
  const int RpH  = (Rh  + 15) & ~15, RpT  = (Rt  + 15) & ~15;
  const int RpEH = (Reh + 15) & ~15, RpET = (Ret + 15) & ~15;

  const size_t lds_n = lds_bytes(RpH, RpT);
  const size_t lds_e = lds_bytes(RpEH, RpET);
  const size_t lds_max = lds_n > lds_e ? lds_n : lds_e;
  (void)hipFuncSetAttribute(reinterpret_cast<const void*>(mutual_attn_kernel),
                            hipFuncAttributeMaxDynamicSharedMemorySize,
                            (int)lds_max);

  const size_t BD = (size_t)B * D;
  // Node-level mutual attention -> s_h, s_t
  hipLaunchKernelGGL(mutual_attn_kernel, dim3(B), dim3(NTHREADS), lds_n, stream,
                     xnh, xnt, Whw, Whb, Wtw, Wtb,
                     out + 0, out + BD, Rh, Rt, RpH, RpT);
  // Line-graph mutual attention -> s_h_line, s_t_line
  hipLaunchKernelGGL(mutual_attn_kernel, dim3(B), dim3(NTHREADS), lds_e, stream,
                     xlh, xlt, Wlhw, Wlhb, Wltw, Wltb,
                     out + 2 * BD, out + 3 * BD, Reh, Ret, RpEH, RpET);
}